// GraphPolicyGenerator_72112500899903
// MI455X (gfx1250) — compile-verified
//
#include <hip/hip_runtime.h>
#include <math.h>

// ---- problem constants (match reference) ----
#define N_NODES 10000
#define N_EDGES 640000
#define DHID    128     // D_IN == D_HID == 128
#define D_OUT   256
#define NODE_F  (N_NODES * DHID)   // 1,280,000 floats per node-feature buffer

typedef __attribute__((ext_vector_type(2))) float v2f;
typedef __attribute__((ext_vector_type(8))) float v8f;

// ---------------------------------------------------------------------------
// zero scratch (atomic-add targets must start at 0)
// ---------------------------------------------------------------------------
__global__ void zero_f32(float* __restrict__ p, int n) {
    int i = blockIdx.x * blockDim.x + threadIdx.x;
    if (i < n) p[i] = 0.0f;
}

// ---------------------------------------------------------------------------
// scatter-add: out[dst[e], :] += feat[src[e], :]
// one wave32 per edge; each lane handles 4 contiguous floats (float4).
// src[e]/dst[e] are wave-uniform -> scalar-cached loads.
// ---------------------------------------------------------------------------
__global__ __launch_bounds__(256)
void scatter_add128(const float* __restrict__ feat,
                    const int* __restrict__ src,
                    const int* __restrict__ dst,
                    float* __restrict__ out,
                    int nEdges) {
    int gid = blockIdx.x * blockDim.x + threadIdx.x;
    int e = gid >> 5;                 // edge index (one per wave)
    if (e >= nEdges) return;
    int c = (gid & 31) << 2;          // feature offset: lane*4
    int s = src[e];
    int d = dst[e];
    const float4 v = *(const float4*)(feat + (size_t)s * DHID + c);
    float* o = out + (size_t)d * DHID + c;
    atomicAdd(o + 0, v.x);
    atomicAdd(o + 1, v.y);
    atomicAdd(o + 2, v.z);
    atomicAdd(o + 3, v.w);
}

// ---------------------------------------------------------------------------
// Out = relu(A @ W + bias), A:[M,128] row-major, W:[128,128] row-major,
// exact fp32 via V_WMMA_F32_16X16X4_F32 (K=4 per issue, 32 issues / tile).
// grid.x = M/16 row-panels; block = 256 (8 wave32), wave w owns N-tile w.
//
// 32-bit A 16x4 layout: lanes 0-15 -> K={0,1} in v0/v1; lanes 16-31 -> K={2,3}.
// 32-bit B 4x16 layout: VGPR j: lanes 0-15 row K=j, lanes 16-31 row K=j+2.
// 32-bit C/D 16x16:     VGPR v: lanes 0-15 row M=v, lanes 16-31 row M=v+8.
// ---------------------------------------------------------------------------
__global__ __launch_bounds__(256)
void gemm128_wmma_relu(const float* __restrict__ A,
                       const float* __restrict__ W,
                       const float* __restrict__ bias,
                       float* __restrict__ Out) {
    const int lane = threadIdx.x & 31;
    const int wave = threadIdx.x >> 5;          // N-tile 0..7
    const int m0   = blockIdx.x * 16;
    const int mA   = lane & 15;                 // A row within tile
    const int kOff = (lane >> 4) * 2;           // 0 (lanes 0-15) or 2 (16-31)
    const int n    = wave * 16 + (lane & 15);   // output column

    const float* __restrict__ arow = A + (size_t)(m0 + mA) * DHID + kOff;
    const float* __restrict__ wcol = W + (size_t)kOff * DHID + n;

    v8f acc = {0.f, 0.f, 0.f, 0.f, 0.f, 0.f, 0.f, 0.f};

    #pragma unroll 8
    for (int kk = 0; kk < DHID; kk += 4) {
        v2f a, b;
        a.x = arow[kk + 0];                     // A[m][kk+kOff+0]
        a.y = arow[kk + 1];                     // A[m][kk+kOff+1]
        b.x = wcol[(size_t)kk * DHID];          // W[kk+kOff+0][n]
        b.y = wcol[(size_t)kk * DHID + DHID];   // W[kk+kOff+1][n]
        acc = __builtin_amdgcn_wmma_f32_16x16x4_f32(
            /*neg_a=*/false, a, /*neg_b=*/false, b,
            /*c_mod=*/(short)0, acc, /*reuse_a=*/false, /*reuse_b=*/false);
    }

    const float bn = bias[n];
    const int rowHalf = (lane >> 4) * 8;
    #pragma unroll
    for (int v = 0; v < 8; ++v) {
        int row = m0 + v + rowHalf;
        float r = acc[v] + bn;
        Out[(size_t)row * DHID + n] = r > 0.f ? r : 0.f;
    }
}

// ---------------------------------------------------------------------------
// pooled[t] = mean over nodes of h2[:, t]   (partial sums + f32 atomics)
// ---------------------------------------------------------------------------
#define POOL_ROWS 100
__global__ __launch_bounds__(128)
void pool_mean(const float* __restrict__ h2, float* __restrict__ pooled) {
    int t  = threadIdx.x;              // feature 0..127
    int r0 = blockIdx.x * POOL_ROWS;
    float s = 0.f;
    #pragma unroll 4
    for (int r = 0; r < POOL_ROWS; ++r) {
        int row = r0 + r;
        if (row < N_NODES) s += h2[(size_t)row * DHID + t];
    }
    atomicAdd(&pooled[t], s * (1.0f / (float)N_NODES));
}

// ---------------------------------------------------------------------------
// out = sigmoid(relu(pooled @ dw1 + db1) @ dw2 + db2)  -- single block
// ---------------------------------------------------------------------------
__global__ __launch_bounds__(256)
void mlp_head(const float* __restrict__ pooled,
              const float* __restrict__ dw1, const float* __restrict__ db1,
              const float* __restrict__ dw2, const float* __restrict__ db2,
              float* __restrict__ out) {
    __shared__ float pl[DHID];
    __shared__ float dl[DHID];
    int t = threadIdx.x;
    if (t < DHID) pl[t] = pooled[t];
    __syncthreads();
    if (t < DHID) {
        float s = db1[t];
        #pragma unroll 8
        for (int k = 0; k < DHID; ++k) s += pl[k] * dw1[k * DHID + t];
        dl[t] = s > 0.f ? s : 0.f;
    }
    __syncthreads();
    float s = db2[t];
    #pragma unroll 8
    for (int k = 0; k < DHID; ++k) s += dl[k] * dw2[k * D_OUT + t];
    out[t] = 1.0f / (1.0f + __expf(-s));
}

// ---------------------------------------------------------------------------
extern "C" void kernel_launch(void* const* d_in, const int* in_sizes, int n_in,
                              void* d_out, int out_size, void* d_ws, size_t ws_size,
                              hipStream_t stream) {
    const float* x   = (const float*)d_in[0];
    const int*   ei  = (const int*)d_in[1];        // edge_index [2, N_EDGES]
    const int*   src = ei;
    const int*   dst = ei + N_EDGES;
    const float* w1  = (const float*)d_in[2];
    const float* b1  = (const float*)d_in[3];
    const float* w2  = (const float*)d_in[4];
    const float* b2  = (const float*)d_in[5];
    const float* dw1 = (const float*)d_in[6];
    const float* db1 = (const float*)d_in[7];
    const float* dw2 = (const float*)d_in[8];
    const float* db2 = (const float*)d_in[9];
    float* out = (float*)d_out;

    // workspace layout (floats): buf0 = agg1 (later reused as h2),
    // buf1 = h, buf2 = agg2, pooled = [128]
    float* buf0   = (float*)d_ws;
    float* buf1   = buf0 + NODE_F;
    float* buf2   = buf1 + NODE_F;
    float* pooled = buf2 + NODE_F;

    // zero all atomic-add targets (agg1, h spans don't hurt, agg2, pooled)
    const int nz = 3 * NODE_F + DHID;
    zero_f32<<<(nz + 255) / 256, 256, 0, stream>>>(buf0, nz);

    const int scatterThreads = N_EDGES * 32;
    const int scatterBlocks  = (scatterThreads + 255) / 256;

    // conv1: agg1 = segsum(x[src] -> dst); h = relu(agg1 @ w1 + b1)
    scatter_add128<<<scatterBlocks, 256, 0, stream>>>(x, src, dst, buf0, N_EDGES);
    gemm128_wmma_relu<<<N_NODES / 16, 256, 0, stream>>>(buf0, w1, b1, buf1);

    // conv2: agg2 = segsum(h[src] -> dst); h2 = relu(agg2 @ w2 + b2)
    scatter_add128<<<scatterBlocks, 256, 0, stream>>>(buf1, src, dst, buf2, N_EDGES);
    gemm128_wmma_relu<<<N_NODES / 16, 256, 0, stream>>>(buf2, w2, b2, buf0);

    // pool + MLP head
    pool_mean<<<(N_NODES + POOL_ROWS - 1) / POOL_ROWS, 128, 0, stream>>>(buf0, pooled);
    mlp_head<<<1, 256, 0, stream>>>(pooled, dw1, db1, dw2, db2, out);
}